// TSRegister_27135603376573
// MI455X (gfx1250) — compile-verified
//
#include <hip/hip_runtime.h>
#include <math.h>

typedef __attribute__((ext_vector_type(2))) float v2f;
typedef __attribute__((ext_vector_type(8))) float v8f;
typedef __attribute__((ext_vector_type(4))) unsigned int u32x4;
typedef __attribute__((ext_vector_type(8))) int i32x8;
typedef __attribute__((ext_vector_type(4))) int i32x4;

#ifndef __has_builtin
#define __has_builtin(x) 0
#endif
#if defined(__HIP_DEVICE_COMPILE__)
#if __has_builtin(__builtin_amdgcn_tensor_load_to_lds)
#define USE_TDM 1
#else
#define USE_TDM 0
#endif
#else
#define USE_TDM 0
#endif
#if __has_include(<hip/amd_detail/amd_gfx1250_TDM.h>)
#define TDM_ARITY6 1
#else
#define TDM_ARITY6 0
#endif

#define D 512
#define R 128
#define TOPK 3
#define TM 16
#define SX_STRIDE 516   // 512 data + 4 pad dwords per row (TDM: 2 @ d=256, 2 @ end)
#define SW_STRIDE 18    // 16 data + 2 pad dwords per row; 18n mod 64 distinct for n<16
#define SS_STRIDE 132
#define SW_SLAB   (D * SW_STRIDE)           // 9216 floats per W-slab buffer

// LDS layout (floats):
#define OFF_SX    0                          // 16*516 = 8256 (overlaid by sims later)
#define OFF_SXE   (OFF_SX + TM * SX_STRIDE)  // 8256
#define OFF_SW    (OFF_SXE + TM * SX_STRIDE) // 2 * 9216 = 18432 (double buffered)
#define OFF_PART  (OFF_SW + 2 * SW_SLAB)     // 256
#define OFF_NX    (OFF_PART + 256)           // 16
#define OFF_TOPV  (OFF_NX + 16)              // 48
#define OFF_TOPI  (OFF_TOPV + 48)            // 48 (int)
#define SMEM_FLOATS (OFF_TOPI + 48)
#define SMEM_BYTES  (SMEM_FLOATS * 4)

#if USE_TDM
// Issue one 2D TDM tile load (global -> LDS) with LDS padding.
// D# packing per CDNA5 ISA.
__device__ __forceinline__ void tdm_load_2d(
    unsigned lds_byte_off, const void* gptr,
    unsigned tensor_d0, unsigned tensor_d1, unsigned stride0,
    unsigned tile_d0, unsigned tile_d1,
    unsigned pad_i_code, unsigned pad_a_code)
{
    unsigned long long ga = (unsigned long long)(uintptr_t)gptr;
    u32x4 g0;
    g0.x = 1u;                               // count=1, user mode, no gather
    g0.y = lds_byte_off;                     // [63:32] lds_addr
    g0.z = (unsigned)ga;                     // [95:64] global_addr lo
    g0.w = (unsigned)((ga >> 32) & 0x01ffffffull) | (2u << 30); // addr hi + type=2
    i32x8 g1;
    g1[0] = (int)((2u << 16)                 // data_size = 4B
                | (1u << 20)                 // pad_enable
                | (pad_i_code << 22)
                | (pad_a_code << 25));
    g1[1] = (int)((tensor_d0 & 0xffffu) << 16);               // dim0 [79:48] lo16
    g1[2] = (int)((tensor_d0 >> 16) | ((tensor_d1 & 0xffffu) << 16));
    g1[3] = (int)((tensor_d1 >> 16) | (tile_d0 << 16));       // tile_dim0 [127:112]
    g1[4] = (int)(tile_d1 & 0xffffu);                          // tile_dim1, tile_dim2=0
    g1[5] = (int)stride0;                                      // tensor_dim0_stride lo
    g1[6] = 0;
    g1[7] = 0;
    i32x4 z4 = {0, 0, 0, 0};
#if TDM_ARITY6
    i32x8 z8 = {0, 0, 0, 0, 0, 0, 0, 0};
    __builtin_amdgcn_tensor_load_to_lds(g0, g1, z4, z4, z8, 0);
#else
    __builtin_amdgcn_tensor_load_to_lds(g0, g1, z4, z4, 0);
#endif
}
#endif

// x-tile index: TDM inserts 2 pad dwords every 256 dwords -> mid-row shift.
__device__ __forceinline__ int sx_idx(int m, int d) {
#if USE_TDM
    return m * SX_STRIDE + d + ((d >> 8) << 1);
#else
    return m * SX_STRIDE + d;
#endif
}

__global__ void tsreg_rnorm_kernel(const float* __restrict__ reg, float* __restrict__ rn) {
    int r = threadIdx.x;
    if (r < R) {
        float s = 0.f;
        #pragma unroll 8
        for (int d = 0; d < D; ++d) { float t = reg[r * D + d]; s += t * t; }
        rn[r] = sqrtf(s);
    }
}

__global__ __launch_bounds__(256)
void tsreg_fused_kernel(const float* __restrict__ x,
                        const float* __restrict__ reg,
                        const float* __restrict__ Wm,
                        const float* __restrict__ bvec,
                        const float* __restrict__ u,
                        const float* __restrict__ v,
                        const float* __restrict__ rn,
                        float* __restrict__ out,
                        int B) {
    extern __shared__ float smem[];
    float* sX    = smem + OFF_SX;
    float* sXe   = smem + OFF_SXE;
    float* sW    = smem + OFF_SW;
    float* sPart = smem + OFF_PART;
    float* sNx   = smem + OFF_NX;
    float* sTopV = smem + OFF_TOPV;
    int*   sTopI = (int*)(smem + OFF_TOPI);
    float* sSims = smem + OFF_SX;   // overlays sX (dead after GEMM1)

    const int tid  = threadIdx.x;
    const int lane = tid & 31;
    const int wave = tid >> 5;          // 0..7
    const int half = (lane >> 4);       // 0 or 1
    const int l16  = lane & 15;
    const size_t rowbase = (size_t)blockIdx.x * TM;

#if USE_TDM
    const unsigned lds_base = __builtin_amdgcn_groupstaticsize();
    const unsigned sx_off  = lds_base + OFF_SX * 4u;
    const unsigned sw_off0 = lds_base + OFF_SW * 4u;
    const unsigned sw_off1 = lds_base + (OFF_SW + SW_SLAB) * 4u;

    // Prologue: async DMA of x tile [16x512] (pad 2dw per 256dw -> stride 516)
    // and W slab 0 [512 rows x 16 k] (pad 2dw per 16dw -> stride 18).
    if (wave == 0) {
        tdm_load_2d(sx_off, x + rowbase * D, D, TM, D, D, TM, /*padi=*/7, /*pada=*/1);
        tdm_load_2d(sw_off0, Wm, D, D, D, 16, D, /*padi=*/3, /*pada=*/1);
    }
#else
    // Manual x-tile staging (coalesced)
    #pragma unroll 4
    for (int i = 0; i < (TM * D) / 256; ++i) {
        int e = i * 256 + tid;
        int m = e >> 9;
        int d = e & (D - 1);
        sX[m * SX_STRIDE + d] = x[(rowbase + m) * D + d];
    }
    __syncthreads();
#endif

    // ---- GEMM1: xe = x @ W^T (M=16, N=512, K=512), fp32 WMMA 16x16x4 ----
    v8f acc1[4] = {};
    for (int c = 0; c < D / 16; ++c) {
        const int ks = c * 16;
#if USE_TDM
        if (wave == 0) {
            if (c + 1 < D / 16) {
                tdm_load_2d(((c + 1) & 1) ? sw_off1 : sw_off0,
                            Wm + (ks + 16), D, D, D, 16, D, 3, 1);
                __builtin_amdgcn_s_wait_tensorcnt(1);
            } else {
                __builtin_amdgcn_s_wait_tensorcnt(0);
            }
        }
        __syncthreads();
        const float* sWb = sW + (c & 1) * SW_SLAB;
#else
        #pragma unroll 4
        for (int i = 0; i < 32; ++i) {
            int e = i * 256 + tid;
            int n = e >> 4;
            int kk = e & 15;
            sW[n * SW_STRIDE + kk] = Wm[n * D + ks + kk];
        }
        __syncthreads();
        const float* sWb = sW;
#endif
        #pragma unroll
        for (int kk = 0; kk < 16; kk += 4) {
            v2f a = *(const v2f*)(sX + sx_idx(l16, ks + kk + (half << 1)));
            #pragma unroll
            for (int t = 0; t < 4; ++t) {
                int bn = (wave * 4 + t) * 16 + l16;
                v2f bf = *(const v2f*)(sWb + bn * SW_STRIDE + kk + (half << 1));
                acc1[t] = __builtin_amdgcn_wmma_f32_16x16x4_f32(
                    false, a, false, bf, (short)0, acc1[t], false, false);
            }
        }
        __syncthreads();
    }

#if USE_TDM
    // Issue register slab 0 early (overlaps bias/norm phase).
    if (wave == 0)
        tdm_load_2d(sw_off0, reg, D, R, D, 16, R, 3, 1);
#endif

    // ---- bias add + store xe tile to LDS ----
    #pragma unroll
    for (int t = 0; t < 4; ++t) {
        int col = (wave * 4 + t) * 16 + l16;
        float bc = bvec[col];
        int rb = half << 3;
        #pragma unroll
        for (int r = 0; r < 8; ++r)
            sXe[(rb + r) * SX_STRIDE + col] = acc1[t][r] + bc;
    }
    __syncthreads();

    // ---- row norms of xe ----
    {
        int m = tid >> 4, seg = tid & 15;
        const float* rp = sXe + m * SX_STRIDE + seg * 32;
        float s = 0.f;
        #pragma unroll 8
        for (int i = 0; i < 32; ++i) { float t = rp[i]; s += t * t; }
        sPart[m * 16 + seg] = s;
    }
    __syncthreads();
    if (tid < TM) {
        float s = 0.f;
        #pragma unroll
        for (int i = 0; i < 16; ++i) s += sPart[tid * 16 + i];
        sNx[tid] = sqrtf(s);
    }
    __syncthreads();

    // ---- GEMM2: sims = xe @ reg^T (M=16, N=128, K=512), one N-tile per wave ----
    v8f acc2 = {};
    for (int c = 0; c < D / 16; ++c) {
        const int ks = c * 16;
#if USE_TDM
        if (wave == 0) {
            if (c + 1 < D / 16) {
                tdm_load_2d(((c + 1) & 1) ? sw_off1 : sw_off0,
                            reg + (ks + 16), D, R, D, 16, R, 3, 1);
                __builtin_amdgcn_s_wait_tensorcnt(1);
            } else {
                __builtin_amdgcn_s_wait_tensorcnt(0);
            }
        }
        __syncthreads();
        const float* sWb = sW + (c & 1) * SW_SLAB;
#else
        #pragma unroll
        for (int i = 0; i < 8; ++i) {
            int e = i * 256 + tid;
            int r = e >> 4;
            int kk = e & 15;
            sW[r * SW_STRIDE + kk] = reg[r * D + ks + kk];
        }
        __syncthreads();
        const float* sWb = sW;
#endif
        #pragma unroll
        for (int kk = 0; kk < 16; kk += 4) {
            v2f a = *(const v2f*)(sXe + l16 * SX_STRIDE + ks + kk + (half << 1));
            v2f bf = *(const v2f*)(sWb + (wave * 16 + l16) * SW_STRIDE + kk + (half << 1));
            acc2 = __builtin_amdgcn_wmma_f32_16x16x4_f32(
                false, a, false, bf, (short)0, acc2, false, false);
        }
        __syncthreads();
    }

    // ---- cosine scale + write sims tile to LDS ----
    {
        int col = wave * 16 + l16;           // register index 0..127
        float rc = rn[col];
        int rb = half << 3;
        #pragma unroll
        for (int r = 0; r < 8; ++r) {
            int row = rb + r;
            float s = acc2[r] / fmaxf(sNx[row] * rc, 1e-8f);
            sSims[row * SS_STRIDE + col] = s;
        }
    }
    __syncthreads();

    // ---- top-3 per row (stable: strict > keeps earliest index on ties) ----
    if (tid < TM) {
        float v0 = -3.4e38f, v1 = -3.4e38f, v2 = -3.4e38f;
        int i0 = 0, i1 = 0, i2 = 0;
        for (int r = 0; r < R; ++r) {
            float s = sSims[tid * SS_STRIDE + r];
            if (s > v0)      { v2 = v1; i2 = i1; v1 = v0; i1 = i0; v0 = s; i0 = r; }
            else if (s > v1) { v2 = v1; i2 = i1; v1 = s; i1 = r; }
            else if (s > v2) { v2 = s; i2 = r; }
        }
        sTopV[tid * 3 + 0] = v0; sTopV[tid * 3 + 1] = v1; sTopV[tid * 3 + 2] = v2;
        sTopI[tid * 3 + 0] = i0; sTopI[tid * 3 + 1] = i1; sTopI[tid * 3 + 2] = i2;
    }
    __syncthreads();

    // ---- outputs: adapted centers [B,3,512] (b128 stores), indices, values ----
    float* outC = out;
    float* outI = out + (size_t)B * TOPK * D;
    float* outV = outI + (size_t)B * TOPK;

    const int NV4 = TM * TOPK * (D / 4);     // 6144 float4s per block
    for (int e = tid; e < NV4; e += 256) {
        int i   = e / (TOPK * (D / 4));          // row in tile
        int rem = e - i * (TOPK * (D / 4));
        int j   = rem / (D / 4);                 // 0..2
        int d4  = rem - j * (D / 4);             // 0..127
        int idx = sTopI[i * 3 + j];
        float4 rv = *(const float4*)(reg + idx * D + d4 * 4);
        float4 vv = *(const float4*)(v + d4 * 4);
        float uj = u[j];
        float4 o;
        o.x = rv.x + uj * vv.x;
        o.y = rv.y + uj * vv.y;
        o.z = rv.z + uj * vv.z;
        o.w = rv.w + uj * vv.w;
        *(float4*)(outC + (rowbase + i) * (TOPK * D) + j * D + d4 * 4) = o;
    }
    if (tid < TM * TOPK) {
        int i = tid / 3, j = tid - i * 3;
        outI[(rowbase + i) * 3 + j] = (float)sTopI[i * 3 + j];
        outV[(rowbase + i) * 3 + j] = sTopV[i * 3 + j];
    }
}

extern "C" void kernel_launch(void* const* d_in, const int* in_sizes, int n_in,
                              void* d_out, int out_size, void* d_ws, size_t ws_size,
                              hipStream_t stream) {
    (void)n_in; (void)out_size; (void)ws_size;
    const float* x   = (const float*)d_in[0];
    const float* reg = (const float*)d_in[1];
    const float* Wm  = (const float*)d_in[2];
    const float* b   = (const float*)d_in[3];
    const float* u   = (const float*)d_in[4];
    const float* v   = (const float*)d_in[5];
    const int B = in_sizes[0] / D;      // 131072

    float* rn = (float*)d_ws;           // 128 floats of workspace
    tsreg_rnorm_kernel<<<1, 128, 0, stream>>>(reg, rn);

    dim3 grid(B / TM);
    tsreg_fused_kernel<<<grid, 256, SMEM_BYTES, stream>>>(
        x, reg, Wm, b, u, v, rn, (float*)d_out, B);
}